// RGB_HVI_25168508354963
// MI455X (gfx1250) — compile-verified
//
#include <hip/hip_runtime.h>
#include <math.h>

#define PI_F  3.14159265358979323846f
#define EPS_F 1e-8f
#define HW    (1024 * 1024)   // per-plane pixel count (1024x1024)

typedef float v4f __attribute__((ext_vector_type(4)));  // native vector: OK for nontemporal builtins

__device__ __forceinline__ float clampf(float x, float lo, float hi) {
    return fminf(fmaxf(x, lo), hi);
}

// base > 0 guaranteed here (sin(v*pi/2)+1e-8, v in [0,1]) -> skip powf's
// special-case ladders. v_exp_f32 is exp2-native; __builtin_amdgcn_exp2f
// avoids the glibc host __exp2f name collision.
__device__ __forceinline__ float pow_pos(float x, float k) {
    return __builtin_amdgcn_exp2f(k * __log2f(x));
}

// ---------------------------------------------------------------------------
// Kernel 0: reset the workspace max accumulator (harness poisons ws).
// ---------------------------------------------------------------------------
__global__ void hvi_init_ws(unsigned int* __restrict__ wsmax) {
    if (blockIdx.x == 0 && threadIdx.x == 0) wsmax[0] = 0u;  // bits of +0.0f
}

// ---------------------------------------------------------------------------
// Kernel 1: global max of val = max(r,g,b). Streams img with regular-temporal
// loads (warms the 192MB L2 for pass 2). LDS tree + one u32 atomicMax/block
// (float bits are order-preserving for vals >= 0 -> deterministic).
// ---------------------------------------------------------------------------
__global__ void hvi_reduce_max(const float* __restrict__ img,
                               unsigned int* __restrict__ wsmax,
                               int ngroups) {
    __shared__ float sm[256];
    float lm = 0.0f;
    const int stride = gridDim.x * blockDim.x;
    for (int g = blockIdx.x * blockDim.x + threadIdx.x; g < ngroups; g += stride) {
        const int p     = g << 2;
        const int batch = p / HW;
        const int rem   = p - batch * HW;
        const float* base = img + (size_t)batch * (3 * HW) + rem;

        const v4f r4 = *(const v4f*)(base);
        const v4f g4 = *(const v4f*)(base + HW);
        const v4f b4 = *(const v4f*)(base + 2 * HW);

        // stay ahead of HBM latency (global_prefetch_b8), keep lines cached
        __builtin_prefetch(base + (size_t)stride * 4,          0, 3);
        __builtin_prefetch(base + HW + (size_t)stride * 4,     0, 3);
        __builtin_prefetch(base + 2 * HW + (size_t)stride * 4, 0, 3);

        float m0 = fmaxf(r4.x, fmaxf(g4.x, b4.x));
        float m1 = fmaxf(r4.y, fmaxf(g4.y, b4.y));
        float m2 = fmaxf(r4.z, fmaxf(g4.z, b4.z));
        float m3 = fmaxf(r4.w, fmaxf(g4.w, b4.w));
        lm = fmaxf(lm, fmaxf(fmaxf(m0, m1), fmaxf(m2, m3)));
    }
    sm[threadIdx.x] = lm;
    __syncthreads();
    for (int s = 128; s > 0; s >>= 1) {
        if (threadIdx.x < s) sm[threadIdx.x] = fmaxf(sm[threadIdx.x], sm[threadIdx.x + s]);
        __syncthreads();
    }
    if (threadIdx.x == 0) atomicMax(wsmax, __float_as_uint(sm[0]));
}

// ---------------------------------------------------------------------------
// Fused per-pixel math: _hvit followed by _phvit. Fully branchless; native
// v_sin/v_cos/v_rcp/v_log/v_exp for the bounded-argument transcendentals.
// ---------------------------------------------------------------------------
__device__ __forceinline__ void hvi_pixel(float rr, float gg, float bb, float e,
                                          float dk, float invM1, float invM2,
                                          float& ro, float& go, float& bo) {
    // ---- _hvit ----
    const float val = fmaxf(rr, fmaxf(gg, bb));
    const float mn  = fminf(rr, fminf(gg, bb));
    const float d   = val - mn + EPS_F;
    const float rd  = __fdividef(1.0f, d);           // v_rcp_f32 path

    // (g-b)/d in (-1,1) -> JAX floor-mod-6 is just (+6 if negative)
    float hr = (gg - bb) * rd;
    hr = (hr < 0.0f) ? hr + 6.0f : hr;
    const float hg = 2.0f + (bb - rr) * rd;
    const float hb = 4.0f + (rr - gg) * rd;
    // where-chain priority: (min==val) > (r==val) > (g==val) > (b==val)
    float hue = (mn == val) ? 0.0f
              : (rr == val) ? hr
              : (gg == val) ? hg
              :               hb;
    hue *= (1.0f / 6.0f);

    const float sat = (val == 0.0f) ? 0.0f : (val - mn) * __fdividef(1.0f, val + EPS_F);
    const float E   = clampf(e, 0.0f, 1.0f);
    const float es  = (E == 1.0f) ? 0.5f : 0.2f;

    // sin(val*pi/2), val in [0,1] -> hardware v_sin_f32 range is trivial here
    const float sv  = __sinf(val * (0.5f * PI_F));

    const float k1  = 0.8f * dk + 0.1f * (val * invM1) + 0.1f * es;
    const float cs1 = pow_pos(sv + EPS_F, k1);

    const float ang = (2.0f * PI_F) * hue;           // in [0, 2pi)
    const float X = cs1 * sat * __cosf(ang);
    const float Y = cs1 * sat * __sinf(ang);
    const float Z = val;

    // ---- _phvit ----
    float Hc = clampf(X, -1.0f, 1.0f);
    float Vc = clampf(Y, -1.0f, 1.0f);
    const float I = clampf(Z, 0.0f, 1.0f);

    const float k2   = 0.8f * dk + 0.1f * (I * invM2) + 0.1f * es;
    const float cs2  = pow_pos(__sinf(I * (0.5f * PI_F)) + EPS_F, k2);
    const float ics2 = __fdividef(1.0f, cs2 + EPS_F);
    Hc = clampf(Hc * ics2, -1.0f, 1.0f);
    Vc = clampf(Vc * ics2, -1.0f, 1.0f);

    // keep accurate atan2f: its ulps gate the floor(6h) sextant select below
    float h = atan2f(Vc, Hc) * (1.0f / (2.0f * PI_F));
    h -= floorf(h);                                  // floor-mod 1.0
    const float s = clampf(sqrtf(Hc * Hc + Vc * Vc + EPS_F), 0.0f, 1.0f);
    const float v = I;

    const float h6  = h * 6.0f;
    const float hif = floorf(h6);
    const float f   = h6 - hif;
    const float pp  = v * (1.0f - s);
    const float qq  = v * (1.0f - f * s);
    const float tt  = v * (1.0f - (1.0f - f) * s);

    const bool c0 = (hif == 0.0f), c1 = (hif == 1.0f), c2 = (hif == 2.0f);
    const bool c3 = (hif == 3.0f), c4 = (hif == 4.0f), c5 = (hif == 5.0f);
    // jnp.select columns, default 0 -> v_cndmask chains, no branches
    ro = c0 ? v  : c1 ? qq : c2 ? pp : c3 ? pp : c4 ? tt : c5 ? v  : 0.0f;
    go = c0 ? tt : c1 ? v  : c2 ? v  : c3 ? qq : c4 ? pp : c5 ? pp : 0.0f;
    bo = c0 ? pp : c1 ? pp : c2 ? tt : c3 ? v  : c4 ? v  : c5 ? qq : 0.0f;
}

// ---------------------------------------------------------------------------
// Kernel 2: fused transform. One v4f group (4 pixels) per thread.
// img loads regular-temporal (expect L2 hits from pass 1); edge load and
// all output stores non-temporal (read/written once) to protect L2 residency.
// ---------------------------------------------------------------------------
__global__ void hvi_transform(const float* __restrict__ img,
                              const float* __restrict__ edge,
                              const float* __restrict__ dkp,
                              const unsigned int* __restrict__ wsmax,
                              float* __restrict__ out,
                              int ngroups) {
    const int g = blockIdx.x * blockDim.x + threadIdx.x;
    if (g >= ngroups) return;

    const float dk    = dkp[0];
    const float M1    = __uint_as_float(wsmax[0]);      // max(val)
    const float M2    = clampf(M1, 0.0f, 1.0f);         // max(clip(val,0,1))
    const float invM1 = 1.0f / (M1 + EPS_F);
    const float invM2 = 1.0f / (M2 + EPS_F);

    const int p     = g << 2;
    const int batch = p / HW;
    const int rem   = p - batch * HW;

    const float* ibase = img  + (size_t)batch * (3 * HW) + rem;
    const float* ebase = edge + (size_t)batch * HW + rem;
    float*       obase = out  + (size_t)batch * (3 * HW) + rem;

    const v4f r4 = *(const v4f*)(ibase);
    const v4f g4 = *(const v4f*)(ibase + HW);
    const v4f b4 = *(const v4f*)(ibase + 2 * HW);
    const v4f e4 = __builtin_nontemporal_load((const v4f*)ebase);

    float ri[4] = {r4.x, r4.y, r4.z, r4.w};
    float gi[4] = {g4.x, g4.y, g4.z, g4.w};
    float bi[4] = {b4.x, b4.y, b4.z, b4.w};
    float ei[4] = {e4.x, e4.y, e4.z, e4.w};
    float rv[4], gv[4], bv[4];

#pragma unroll
    for (int i = 0; i < 4; ++i) {
        hvi_pixel(ri[i], gi[i], bi[i], ei[i], dk, invM1, invM2, rv[i], gv[i], bv[i]);
    }

    v4f rout = {rv[0], rv[1], rv[2], rv[3]};
    v4f gout = {gv[0], gv[1], gv[2], gv[3]};
    v4f bout = {bv[0], bv[1], bv[2], bv[3]};
    __builtin_nontemporal_store(rout, (v4f*)(obase));
    __builtin_nontemporal_store(gout, (v4f*)(obase + HW));
    __builtin_nontemporal_store(bout, (v4f*)(obase + 2 * HW));
}

// ---------------------------------------------------------------------------
// Launch: init ws -> global-max reduce -> fused transform (all on `stream`).
// ---------------------------------------------------------------------------
extern "C" void kernel_launch(void* const* d_in, const int* in_sizes, int n_in,
                              void* d_out, int out_size, void* d_ws, size_t ws_size,
                              hipStream_t stream) {
    const float* img  = (const float*)d_in[0];   // (8,3,1024,1024) fp32
    const float* edge = (const float*)d_in[1];   // (8,1,1024,1024) fp32
    const float* dk   = (const float*)d_in[2];   // (1,) fp32
    unsigned int* wsmax = (unsigned int*)d_ws;
    float* out = (float*)d_out;                  // (8,3,1024,1024) fp32

    const int total_pixels = in_sizes[1];        // B * HW = 8 * 1048576
    const int ngroups      = total_pixels >> 2;  // float4 groups

    hvi_init_ws<<<1, 1, 0, stream>>>(wsmax);

    // Streaming reduction: 1024 blocks x 256 threads (8 wave32s/block),
    // grid-stride so the prefetch distance is well-defined.
    hvi_reduce_max<<<1024, 256, 0, stream>>>(img, wsmax, ngroups);

    const int blocks = (ngroups + 255) / 256;
    hvi_transform<<<blocks, 256, 0, stream>>>(img, edge, dk, wsmax, out, ngroups);
}